// PyG_GCN_3246995276081
// MI455X (gfx1250) — compile-verified
//
#include <hip/hip_runtime.h>

typedef float v2f __attribute__((ext_vector_type(2)));
typedef float v8f __attribute__((ext_vector_type(8)));

#define DIN  128
#define DHID 64
#define DOUT 64

// ---------------- elementwise / degree kernels ----------------

__global__ void k_fill(float* __restrict__ p, float v, int n) {
    int i = blockIdx.x * blockDim.x + threadIdx.x;
    if (i < n) p[i] = v;
}

__global__ void k_degree(const int* __restrict__ dst, float* __restrict__ deg, int nE) {
    int e = blockIdx.x * blockDim.x + threadIdx.x;
    if (e < nE) unsafeAtomicAdd(&deg[dst[e]], 1.0f);
}

__global__ void k_rsqrt_inplace(float* __restrict__ d, int n) {
    int i = blockIdx.x * blockDim.x + threadIdx.x;
    if (i < n) d[i] = rsqrtf(d[i]);   // deg >= 1 always (self-loops)
}

// ---------------- GEMM layer 1: Xl = X @ W1 ; agg = Xl * dinv^2 ----------------
// 128 threads = 4 waves; each wave computes a 16x64 strip via 4 WMMA f32 16x16x4 tiles.

__global__ void __launch_bounds__(128)
k_gemm1(const float* __restrict__ X, const float* __restrict__ W,
        const float* __restrict__ dinv,
        float* __restrict__ Xl, float* __restrict__ agg, int nV) {
    __shared__ float Wl[DIN * DHID];                 // 32 KB
    for (int i = threadIdx.x; i < DIN * DHID; i += 128) Wl[i] = W[i];
    __syncthreads();

    const int wave  = threadIdx.x >> 5;
    const int lane  = threadIdx.x & 31;
    const int m0    = (blockIdx.x * 4 + wave) * 16;  // 16-row tile
    if (m0 >= nV) return;                            // wave-uniform guard: EXEC stays all-1s

    const int mlane = lane & 15;
    const int khalf = lane >> 4;                     // 0 or 1

    v8f acc[4] = {};
    const float* Xrow = X + (size_t)(m0 + mlane) * DIN;

    for (int k0 = 0; k0 < DIN; k0 += 4) {
        const int kb = k0 + khalf * 2;
        v2f a;
        a.x = Xrow[kb];
        a.y = Xrow[kb + 1];
#pragma unroll
        for (int nt = 0; nt < 4; ++nt) {
            v2f b;
            b.x = Wl[kb * DHID       + nt * 16 + mlane];
            b.y = Wl[(kb + 1) * DHID + nt * 16 + mlane];
            acc[nt] = __builtin_amdgcn_wmma_f32_16x16x4_f32(
                false, a, false, b, (short)0, acc[nt], false, false);
        }
    }

    const int roff = 8 * khalf;
#pragma unroll
    for (int r = 0; r < 8; ++r) {
        const int row = m0 + r + roff;
        const float s  = dinv[row];
        const float s2 = s * s;
#pragma unroll
        for (int nt = 0; nt < 4; ++nt) {
            const int col = nt * 16 + mlane;
            const float v = acc[nt][r];
            const size_t idx = (size_t)row * DHID + col;
            Xl[idx]  = v;
            agg[idx] = v * s2;                       // self-loop message seed
        }
    }
}

// ---------------- GEMM layer 2: Hl = relu(agg+b1) @ W2 ; out = Hl*dinv^2 + b2 ----------------

__global__ void __launch_bounds__(128)
k_gemm2(const float* __restrict__ agg, const float* __restrict__ b1,
        const float* __restrict__ W, const float* __restrict__ b2,
        const float* __restrict__ dinv,
        float* __restrict__ Hl, float* __restrict__ out, int nV) {
    __shared__ float Wl[DHID * DOUT];                // 16 KB
    __shared__ float b1l[DHID];
    __shared__ float b2l[DOUT];
    for (int i = threadIdx.x; i < DHID * DOUT; i += 128) Wl[i] = W[i];
    if (threadIdx.x < DHID) b1l[threadIdx.x] = b1[threadIdx.x];
    if (threadIdx.x < DOUT) b2l[threadIdx.x] = b2[threadIdx.x];
    __syncthreads();

    const int wave  = threadIdx.x >> 5;
    const int lane  = threadIdx.x & 31;
    const int m0    = (blockIdx.x * 4 + wave) * 16;
    if (m0 >= nV) return;

    const int mlane = lane & 15;
    const int khalf = lane >> 4;

    v8f acc[4] = {};
    const float* Arow = agg + (size_t)(m0 + mlane) * DHID;

    for (int k0 = 0; k0 < DHID; k0 += 4) {
        const int kb = k0 + khalf * 2;
        v2f a;
        a.x = fmaxf(Arow[kb]     + b1l[kb],     0.0f);   // fused bias + ReLU
        a.y = fmaxf(Arow[kb + 1] + b1l[kb + 1], 0.0f);
#pragma unroll
        for (int nt = 0; nt < 4; ++nt) {
            v2f b;
            b.x = Wl[kb * DOUT       + nt * 16 + mlane];
            b.y = Wl[(kb + 1) * DOUT + nt * 16 + mlane];
            acc[nt] = __builtin_amdgcn_wmma_f32_16x16x4_f32(
                false, a, false, b, (short)0, acc[nt], false, false);
        }
    }

    const int roff = 8 * khalf;
#pragma unroll
    for (int r = 0; r < 8; ++r) {
        const int row = m0 + r + roff;
        const float s  = dinv[row];
        const float s2 = s * s;
#pragma unroll
        for (int nt = 0; nt < 4; ++nt) {
            const int col = nt * 16 + mlane;
            const float v = acc[nt][r];
            const size_t idx = (size_t)row * DOUT + col;
            Hl[idx]  = v;
            out[idx] = v * s2 + b2l[col];            // self-loop seed + output bias
        }
    }
}

// ---------------- edge scatter: out[dst] += F[src] * dinv[src]*dinv[dst] ----------------
// One wave32 per edge; each lane owns 2 of the 64 columns.

__global__ void __launch_bounds__(256)
k_scatter(const int* __restrict__ src, const int* __restrict__ dst,
          const float* __restrict__ dinv, const float* __restrict__ F,
          float* __restrict__ out, int nE) {
    const int e = blockIdx.x * 8 + (threadIdx.x >> 5);
    if (e >= nE) return;
    const int lane = threadIdx.x & 31;
    const int s = src[e];
    const int d = dst[e];
    const float w = dinv[s] * dinv[d];
    const float2 v = *(const float2*)(F + (size_t)s * 64 + lane * 2);
    float* o = out + (size_t)d * 64 + lane * 2;
    unsafeAtomicAdd(o,     v.x * w);
    unsafeAtomicAdd(o + 1, v.y * w);
}

// ---------------- launch ----------------

extern "C" void kernel_launch(void* const* d_in, const int* in_sizes, int n_in,
                              void* d_out, int out_size, void* d_ws, size_t ws_size,
                              hipStream_t stream) {
    const int*   E   = (const int*)d_in[1];
    const float* X   = (const float*)d_in[2];
    const float* W1  = (const float*)d_in[3];
    const float* b1  = (const float*)d_in[4];
    const float* W2  = (const float*)d_in[5];
    const float* b2  = (const float*)d_in[6];

    const int nE = in_sizes[1] / 2;
    const int nV = in_sizes[2] / DIN;

    const int* src = E;
    const int* dst = E + nE;

    float* ws   = (float*)d_ws;
    float* dinv = ws;                                // nV
    float* Xl   = dinv + nV;                         // nV*64
    float* agg  = Xl + (size_t)nV * DHID;            // nV*64
    float* Hl   = agg + (size_t)nV * DHID;           // nV*64
    float* out  = (float*)d_out;

    const int vb = (nV + 255) / 256;
    const int eb = (nE + 255) / 256;
    const int gb = (nV + 63) / 64;                   // 4 waves (16 rows each) per block
    const int sb = (nE + 7) / 8;                     // 8 edges per 256-thread block

    k_fill<<<vb, 256, 0, stream>>>(dinv, 1.0f, nV);                  // self-loop degree
    k_degree<<<eb, 256, 0, stream>>>(dst, dinv, nE);
    k_rsqrt_inplace<<<vb, 256, 0, stream>>>(dinv, nV);

    k_gemm1<<<gb, 128, 0, stream>>>(X, W1, dinv, Xl, agg, nV);
    k_scatter<<<sb, 256, 0, stream>>>(src, dst, dinv, Xl, agg, nE);

    k_gemm2<<<gb, 128, 0, stream>>>(agg, b1, W2, b2, dinv, Hl, out, nV);
    k_scatter<<<sb, 256, 0, stream>>>(src, dst, dinv, Hl, out, nE);
}